// mv_embedding_2740189135213
// MI455X (gfx1250) — compile-verified
//
#include <hip/hip_runtime.h>

// Problem constants (from reference)
#define B_    1024
#define K0_   256
#define D_    1000            // = 250 float4 per row, 4000 B (16B aligned stride)
#define EMB_  3
#define OUT_  512
#define HID_  (EMB_ * K0_)    // 768
#define ROWS_ (B_ * K0_)      // 262144
#define NF4_  (D_ / 4)        // 250 float4 per row

typedef float v2f __attribute__((ext_vector_type(2)));
typedef float v4f __attribute__((ext_vector_type(4)));
typedef float v8f __attribute__((ext_vector_type(8)));

// ---------------------------------------------------------------------------
// Stage 1: y[row, e] = dot(x[row, :], W_emb[e, :])   row = b*K0 + k
// HBM-bound: 1.05 GB of x streamed once. Weights (3 x 4 KB) held in registers
// per lane; x loads are non-temporal b128 to avoid polluting L2.
// One wave per row iteration; 3 accumulators wave-reduced via shfl_xor.
// ---------------------------------------------------------------------------
__global__ __launch_bounds__(256) void emb_stage1_kernel(
    const float* __restrict__ x,
    const float* __restrict__ Wemb,     // [3, 1000]
    float* __restrict__ Y,              // [ROWS_, 3] == [B, 768] row-major
    int total_waves)
{
  const int lane = threadIdx.x & 31;
  const int wave = blockIdx.x * (blockDim.x >> 5) + (threadIdx.x >> 5);

  // Per-lane fixed float4 positions: pos = lane + 32*j, j = 0..7 (pos < 250)
  v4f w0[8], w1[8], w2[8];
#pragma unroll
  for (int j = 0; j < 8; ++j) {
    const int pos = lane + 32 * j;
    if (pos < NF4_) {
      w0[j] = ((const v4f*)(Wemb + 0 * D_))[pos];
      w1[j] = ((const v4f*)(Wemb + 1 * D_))[pos];
      w2[j] = ((const v4f*)(Wemb + 2 * D_))[pos];
    } else {
      w0[j] = (v4f){0.f, 0.f, 0.f, 0.f};
      w1[j] = (v4f){0.f, 0.f, 0.f, 0.f};
      w2[j] = (v4f){0.f, 0.f, 0.f, 0.f};
    }
  }

  for (int row = wave; row < ROWS_; row += total_waves) {
    const v4f* __restrict__ xr = (const v4f*)(x + (size_t)row * D_);
    float s0 = 0.f, s1 = 0.f, s2 = 0.f;
#pragma unroll
    for (int j = 0; j < 8; ++j) {
      const int pos = lane + 32 * j;
      if (pos < NF4_) {
        v4f v = __builtin_nontemporal_load(&xr[pos]);   // stream x, NT hint
        s0 = fmaf(v.x, w0[j].x, s0); s0 = fmaf(v.y, w0[j].y, s0);
        s0 = fmaf(v.z, w0[j].z, s0); s0 = fmaf(v.w, w0[j].w, s0);
        s1 = fmaf(v.x, w1[j].x, s1); s1 = fmaf(v.y, w1[j].y, s1);
        s1 = fmaf(v.z, w1[j].z, s1); s1 = fmaf(v.w, w1[j].w, s1);
        s2 = fmaf(v.x, w2[j].x, s2); s2 = fmaf(v.y, w2[j].y, s2);
        s2 = fmaf(v.z, w2[j].z, s2); s2 = fmaf(v.w, w2[j].w, s2);
      }
    }
    // wave32 tree reduction
#pragma unroll
    for (int off = 16; off > 0; off >>= 1) {
      s0 += __shfl_xor(s0, off, 32);
      s1 += __shfl_xor(s1, off, 32);
      s2 += __shfl_xor(s2, off, 32);
    }
    if (lane == 0) {
      float* yo = Y + (size_t)row * EMB_;
      yo[0] = s0; yo[1] = s1; yo[2] = s2;
    }
  }
}

// ---------------------------------------------------------------------------
// Stage 2: out[1024,512] = Y[1024,768] @ W_fc2^T[768,512] + b  via
// V_WMMA_F32_16X16X4_F32. One 16x16 C tile per wave, K stepped by 4.
//
// A (16x4 f32) per-lane layout (ISA 7.12.2): lane = half*16 + r,
//   VGPR0 = A[M=r][K = 2*half],  VGPR1 = A[M=r][K = 2*half + 1]
//   -> contiguous float2 at Y[(m0+r)*768 + k + 2*half]
// B (4x16 f32) mirrored: float2 at W_fc2[(n0+r)*768 + k + 2*half]
// C/D (16x16 f32): acc[i] -> out row m0 + i + 8*half, col n0 + r.
// Uniform control flow per wave -> EXEC all ones for WMMA.
// ---------------------------------------------------------------------------
__global__ __launch_bounds__(256) void fc2_wmma_kernel(
    const float* __restrict__ Y,        // [1024, 768]
    const float* __restrict__ Wfc2,     // [512, 768] row-major
    const float* __restrict__ bias,     // [512]
    float* __restrict__ out)            // [1024, 512]
{
  const int lane = threadIdx.x & 31;
  const int tile = blockIdx.x * (blockDim.x >> 5) + (threadIdx.x >> 5); // 0..2047
  const int m0   = (tile >> 5) << 4;    // 64 M-tiles
  const int n0   = (tile & 31) << 4;    // 32 N-tiles
  const int half = lane >> 4;
  const int r    = lane & 15;

  const float* __restrict__ arow = Y    + (size_t)(m0 + r) * HID_ + 2 * half;
  const float* __restrict__ brow = Wfc2 + (size_t)(n0 + r) * HID_ + 2 * half;

  v8f acc = {0.f, 0.f, 0.f, 0.f, 0.f, 0.f, 0.f, 0.f};
#pragma unroll 4
  for (int k = 0; k < HID_; k += 4) {
    v2f a = *(const v2f*)(arow + k);    // b64 load, 8B aligned
    v2f b = *(const v2f*)(brow + k);    // b64 load, 8B aligned
    acc = __builtin_amdgcn_wmma_f32_16x16x4_f32(
        /*neg_a=*/false, a, /*neg_b=*/false, b,
        /*c_mod=*/(short)0, acc, /*reuse_a=*/false, /*reuse_b=*/false);
  }

  const float bv = bias[n0 + r];
#pragma unroll
  for (int i = 0; i < 8; ++i) {
    out[(size_t)(m0 + i + 8 * half) * OUT_ + (n0 + r)] = acc[i] + bv;
  }
}

// ---------------------------------------------------------------------------
// Host-side launch
// ---------------------------------------------------------------------------
extern "C" void kernel_launch(void* const* d_in, const int* in_sizes, int n_in,
                              void* d_out, int out_size, void* d_ws, size_t ws_size,
                              hipStream_t stream) {
  const float* x    = (const float*)d_in[0];   // [1024, 256, 1000]
  const float* Wemb = (const float*)d_in[1];   // [3, 1000]
  const float* Wfc2 = (const float*)d_in[2];   // [512, 768]
  const float* bfc2 = (const float*)d_in[3];   // [512]
  float* out = (float*)d_out;                  // [1024, 512]
  float* Y   = (float*)d_ws;                   // [1024, 768] = 3 MB scratch

  // Stage 1: 1024 blocks x 8 waves = 8192 persistent waves, 32 rows each.
  const int blocks1 = 1024;
  const int total_waves = blocks1 * (256 / 32);
  emb_stage1_kernel<<<blocks1, 256, 0, stream>>>(x, Wemb, Y, total_waves);

  // Stage 2: 2048 tiles of 16x16, 8 waves/block -> 256 blocks.
  fc2_wmma_kernel<<<256, 256, 0, stream>>>(Y, Wfc2, bfc2, out);
}